// GATModel_14723147890708
// MI455X (gfx1250) — compile-verified
//
#include <hip/hip_runtime.h>
#include <hip/hip_bf16.h>

typedef float v2f __attribute__((ext_vector_type(2)));
typedef float v8f __attribute__((ext_vector_type(8)));

#define IN_CH   128
#define HC      256   // HEADS*OUT_CH
#define HEADS   4
#define OUT_CH  64
#define NEG_SLOPE 0.2f

// ---------- helpers: order-preserving f32 <-> u32 encoding for atomic max ----
__device__ __forceinline__ unsigned enc_f32(float f) {
    unsigned u = __float_as_uint(f);
    return (u & 0x80000000u) ? ~u : (u | 0x80000000u);
}
__device__ __forceinline__ float dec_f32(unsigned u) {
    return __uint_as_float((u & 0x80000000u) ? (u & 0x7fffffffu) : ~u);
}
__device__ __forceinline__ float lrelu(float t) { return t > 0.f ? t : NEG_SLOPE * t; }

// ---------- 0: init out = bias, m_enc = 0 (== -inf under enc), denom = 0 -----
__global__ __launch_bounds__(256) void gat_init(float* __restrict__ out,
                                                const float* __restrict__ bias,
                                                unsigned* __restrict__ m_enc,
                                                float* __restrict__ denom,
                                                long long out_elems, int n_stat) {
    long long i = (long long)blockIdx.x * blockDim.x + threadIdx.x;
    if (i < out_elems) out[i] = bias[(int)(i & (HC - 1))];
    if (i < n_stat) { m_enc[i] = 0u; denom[i] = 0.f; }
}

// ---------- 1: h = x @ W via V_WMMA_F32_16X16X4_F32 --------------------------
// One wave per 16x16 output tile. A: 16x4 f32 (2 VGPRs/lane: lanes 0-15 hold
// K=0,1; lanes 16-31 hold K=2,3). B: 4x16 (v0 = row K=2*half, v1 = K=2*half+1,
// N = lane%16). C/D: vgpr g -> row g + 8*half, col lane%16.
__global__ __launch_bounds__(256) void gat_gemm_wmma(const float* __restrict__ x,
                                                     const float* __restrict__ W,
                                                     float* __restrict__ h,
                                                     int n_nodes, int n_tiles) {
    const int lane = threadIdx.x & 31;
    const int wave = threadIdx.x >> 5;
    const int tile = blockIdx.x * 8 + wave;
    if (tile >= n_tiles) return;                 // wave-uniform: EXEC stays full
    const int tm = tile >> 4;                    // 16 column tiles (HC/16)
    const int tn = tile & 15;
    const int half = lane >> 4;
    const int lrow = lane & 15;

    int arow = tm * 16 + lrow;
    if (arow >= n_nodes) arow = n_nodes - 1;     // clamp loads; stores guarded
    const int col = tn * 16 + lrow;

    const float* __restrict__ xrow = x + (size_t)arow * IN_CH + 2 * half;
    const float* __restrict__ wcol = W + (size_t)(2 * half) * HC + col;

    v8f acc = {};
#pragma unroll
    for (int k = 0; k < IN_CH; k += 4) {
        v2f a = *reinterpret_cast<const v2f*>(xrow + k);   // 8B aligned (k, 2*half even)
        v2f b;
        b.x = wcol[(size_t)k * HC];
        b.y = wcol[(size_t)(k + 1) * HC];
        acc = __builtin_amdgcn_wmma_f32_16x16x4_f32(false, a, false, b,
                                                    (short)0, acc, false, false);
    }

    const int rbase = tm * 16 + 8 * half;
    float* __restrict__ hp = h + (size_t)rbase * HC + col;
#pragma unroll
    for (int g = 0; g < 8; ++g) {
        if (rbase + g < n_nodes) hp[(size_t)g * HC] = acc[g];
    }
}

// ---------- 2: per-node attention logits (wave per node, shuffle reduce) -----
__global__ __launch_bounds__(256) void gat_logits(const float* __restrict__ h,
                                                  const float* __restrict__ att_src,
                                                  const float* __restrict__ att_dst,
                                                  float* __restrict__ a_src,
                                                  float* __restrict__ a_dst,
                                                  int n_nodes) {
    const int lane = threadIdx.x & 31;
    const int node = (int)(((long long)blockIdx.x * blockDim.x + threadIdx.x) >> 5);
    if (node >= n_nodes) return;
    const float* __restrict__ hp = h + (size_t)node * HC;
#pragma unroll
    for (int hd = 0; hd < HEADS; ++hd) {
        float v0 = hp[hd * OUT_CH + lane];
        float v1 = hp[hd * OUT_CH + lane + 32];
        float s = v0 * att_src[hd * OUT_CH + lane] + v1 * att_src[hd * OUT_CH + lane + 32];
        float d = v0 * att_dst[hd * OUT_CH + lane] + v1 * att_dst[hd * OUT_CH + lane + 32];
#pragma unroll
        for (int off = 16; off; off >>= 1) {
            s += __shfl_xor(s, off, 32);
            d += __shfl_xor(d, off, 32);
        }
        if (lane == 0) { a_src[node * HEADS + hd] = s; a_dst[node * HEADS + hd] = d; }
    }
}

// ---------- 3: segment max over incoming edges (thread per edge, 4 heads) ----
__global__ __launch_bounds__(256) void gat_edge_max(const long long* __restrict__ ei,
                                                    const float* __restrict__ a_src,
                                                    const float* __restrict__ a_dst,
                                                    unsigned* __restrict__ m_enc,
                                                    int n_edges, int n_nodes) {
    long long idx = (long long)blockIdx.x * blockDim.x + threadIdx.x;
    long long tot = (long long)n_edges + n_nodes;
    if (idx >= tot) return;
    int s, d;
    if (idx < n_edges) { s = (int)ei[idx]; d = (int)ei[n_edges + idx]; }
    else               { s = d = (int)(idx - n_edges); }        // self loop
#pragma unroll
    for (int hd = 0; hd < HEADS; ++hd) {
        float e = lrelu(a_src[s * HEADS + hd] + a_dst[d * HEADS + hd]);
        atomicMax(&m_enc[d * HEADS + hd], enc_f32(e));
    }
}

// ---------- 4: segment sum of exp(e - m) -------------------------------------
__global__ __launch_bounds__(256) void gat_edge_sum(const long long* __restrict__ ei,
                                                    const float* __restrict__ a_src,
                                                    const float* __restrict__ a_dst,
                                                    const unsigned* __restrict__ m_enc,
                                                    float* __restrict__ denom,
                                                    int n_edges, int n_nodes) {
    long long idx = (long long)blockIdx.x * blockDim.x + threadIdx.x;
    long long tot = (long long)n_edges + n_nodes;
    if (idx >= tot) return;
    int s, d;
    if (idx < n_edges) { s = (int)ei[idx]; d = (int)ei[n_edges + idx]; }
    else               { s = d = (int)(idx - n_edges); }
#pragma unroll
    for (int hd = 0; hd < HEADS; ++hd) {
        float e = lrelu(a_src[s * HEADS + hd] + a_dst[d * HEADS + hd]);
        float m = dec_f32(m_enc[d * HEADS + hd]);
        atomicAdd(&denom[d * HEADS + hd], __expf(e - m));
    }
}

// ---------- 5: weighted aggregation (wave per edge, 256 ch, atomic f32 add) --
__global__ __launch_bounds__(256) void gat_aggregate(const long long* __restrict__ ei,
                                                     const float* __restrict__ h,
                                                     const float* __restrict__ a_src,
                                                     const float* __restrict__ a_dst,
                                                     const unsigned* __restrict__ m_enc,
                                                     const float* __restrict__ denom,
                                                     float* __restrict__ out,
                                                     int n_edges, int n_nodes) {
    const int lane = threadIdx.x & 31;
    long long widx = ((long long)blockIdx.x * blockDim.x + threadIdx.x) >> 5;
    long long tot = (long long)n_edges + n_nodes;
    if (widx >= tot) return;
    int s, d;
    if (widx < n_edges) { s = (int)ei[widx]; d = (int)ei[n_edges + widx]; }
    else                { s = d = (int)(widx - n_edges); }
    float al[HEADS];
#pragma unroll
    for (int hd = 0; hd < HEADS; ++hd) {
        float e = lrelu(a_src[s * HEADS + hd] + a_dst[d * HEADS + hd]);
        float m = dec_f32(m_enc[d * HEADS + hd]);
        al[hd] = __expf(e - m) / (denom[d * HEADS + hd] + 1e-16f);
    }
    const float* __restrict__ hs = h + (size_t)s * HC;
    float* __restrict__ od = out + (size_t)d * HC;
#pragma unroll
    for (int it = 0; it < 8; ++it) {            // 8*32 = 256 channels
        int c = lane + 32 * it;
        atomicAdd(&od[c], hs[c] * al[it >> 1]); // head = c/64 = it/2 (compile-time)
    }
}

// ---------------------------------------------------------------------------
extern "C" void kernel_launch(void* const* d_in, const int* in_sizes, int n_in,
                              void* d_out, int out_size, void* d_ws, size_t ws_size,
                              hipStream_t stream) {
    const float*     x       = (const float*)d_in[0];
    const long long* ei      = (const long long*)d_in[1];   // int64 [2, E]
    const float*     W       = (const float*)d_in[2];
    const float*     att_src = (const float*)d_in[3];
    const float*     att_dst = (const float*)d_in[4];
    const float*     bias    = (const float*)d_in[5];
    float*           out     = (float*)d_out;

    const int n_nodes = in_sizes[0] / IN_CH;
    const int n_edges = in_sizes[1] / 2;

    // workspace layout (fp32/u32 words): h | a_src | a_dst | m_enc | denom
    float*    h      = (float*)d_ws;
    float*    a_src  = h + (size_t)n_nodes * HC;
    float*    a_dst  = a_src + (size_t)n_nodes * HEADS;
    unsigned* m_enc  = (unsigned*)(a_dst + (size_t)n_nodes * HEADS);
    float*    denom  = (float*)(m_enc + (size_t)n_nodes * HEADS);

    const long long out_elems = (long long)n_nodes * HC;
    const int       n_stat    = n_nodes * HEADS;
    const long long tot       = (long long)n_edges + n_nodes;

    {   // init
        int blocks = (int)((out_elems + 255) / 256);
        gat_init<<<blocks, 256, 0, stream>>>(out, bias, m_enc, denom, out_elems, n_stat);
    }
    {   // GEMM h = x @ W (f32 WMMA)
        int tiles_m = (n_nodes + 15) / 16;
        int n_tiles = tiles_m * (HC / 16);
        int blocks  = (n_tiles + 7) / 8;        // 8 waves / block
        gat_gemm_wmma<<<blocks, 256, 0, stream>>>(x, W, h, n_nodes, n_tiles);
    }
    {   // per-node logits
        int blocks = (n_nodes + 7) / 8;
        gat_logits<<<blocks, 256, 0, stream>>>(h, att_src, att_dst, a_src, a_dst, n_nodes);
    }
    {   // segment max
        int blocks = (int)((tot + 255) / 256);
        gat_edge_max<<<blocks, 256, 0, stream>>>(ei, a_src, a_dst, m_enc, n_edges, n_nodes);
    }
    {   // segment sum of exp
        int blocks = (int)((tot + 255) / 256);
        gat_edge_sum<<<blocks, 256, 0, stream>>>(ei, a_src, a_dst, m_enc, denom, n_edges, n_nodes);
    }
    {   // weighted aggregation, wave per edge
        long long threads = tot * 32;
        int blocks = (int)((threads + 255) / 256);
        gat_aggregate<<<blocks, 256, 0, stream>>>(ei, h, a_src, a_dst, m_enc, denom, out,
                                                  n_edges, n_nodes);
    }
}